// ProbAttention_22342419874284
// MI455X (gfx1250) — compile-verified
//
#include <hip/hip_runtime.h>
#include <hip/hip_bf16.h>

// ProbSparse (Informer) attention for MI455X / gfx1250.
// Memory-bound problem (~100MB moved, ~1.1 GFLOP) -> keep fp32, use
// V_WMMA_F32_16X16X4_F32 for the two GEMMs (precision-preserving matrix path).
// v2: K/V chunks staged into LDS with coalesced b128 loads; O accumulators
//     live in registers across the whole key stream; next-chunk prefetch.

typedef __attribute__((ext_vector_type(2))) float v2f;
typedef __attribute__((ext_vector_type(8))) float v8f;

#define B_    2
#define L_    4096
#define H_    8
#define D_    64
#define SK    45        // sample_k
#define NTOP  45        // n_top
#define UP    48        // NTOP padded to 3 row-tiles of 16
#define CHUNK 64        // keys per streaming chunk in kC
#define NCHUNK (L_ / CHUNK)
#define SCALE 0.125f    // 1/sqrt(64)
#define NEG_BIG (-3.0e38f)

// ---------------------------------------------------------------- kernel A
// M[b,h,q] = max_s(q . k_sample) - sum_s(q . k_sample)/L
__global__ void kA_metric(const float* __restrict__ q,
                          const float* __restrict__ k,
                          const int*   __restrict__ samp,
                          float*       __restrict__ Mout) {
  int t = blockIdx.x * blockDim.x + threadIdx.x;   // (b*H + h)*L + qi
  if (t >= B_ * H_ * L_) return;
  int qi = t % L_;
  int h  = (t / L_) % H_;
  int b  = t / (L_ * H_);
  const float* qrow = q + (((size_t)b * L_ + qi) * H_ + h) * D_;
  float mx = NEG_BIG, sm = 0.f;
  for (int s = 0; s < SK; ++s) {
    int kidx = samp[qi * SK + s];
    const float* krow = k + (((size_t)b * L_ + kidx) * H_ + h) * D_;
    float acc = 0.f;
#pragma unroll
    for (int d = 0; d < D_; d += 4) {
      float4 qv = *(const float4*)(qrow + d);
      float4 kv = *(const float4*)(krow + d);
      acc += qv.x * kv.x + qv.y * kv.y + qv.z * kv.z + qv.w * kv.w;
    }
    mx = fmaxf(mx, acc);
    sm += acc;
  }
  Mout[t] = mx - sm / (float)L_;
}

// ---------------------------------------------------------------- kernel B
// top-45 indices per (b,h) via iterative argmax (ties -> lowest index, jax-like)
__global__ void kB_topk(const float* __restrict__ Mmet, int* __restrict__ Mtop) {
  __shared__ float mv[L_];   // 16KB
  __shared__ float rv[256];
  __shared__ int   ri[256];
  int bh  = blockIdx.x;
  int tid = threadIdx.x;
  const float* src = Mmet + (size_t)bh * L_;
  for (int i = tid; i < L_; i += 256) mv[i] = src[i];
  __syncthreads();
  for (int it = 0; it < NTOP; ++it) {
    float bestv = NEG_BIG; int besti = L_;
    for (int i = tid; i < L_; i += 256) {
      float v = mv[i];
      if (v > bestv || (v == bestv && i < besti)) { bestv = v; besti = i; }
    }
    rv[tid] = bestv; ri[tid] = besti;
    __syncthreads();
    for (int s = 128; s > 0; s >>= 1) {
      if (tid < s) {
        float ov = rv[tid + s]; int oi = ri[tid + s];
        if (ov > rv[tid] || (ov == rv[tid] && oi < ri[tid])) { rv[tid] = ov; ri[tid] = oi; }
      }
      __syncthreads();
    }
    if (tid == 0) { Mtop[bh * 64 + it] = ri[0]; mv[ri[0]] = NEG_BIG; }
    __syncthreads();
  }
}

// ---------------------------------------------------------------- kernel C
// Flash-style sparse attention with V_WMMA_F32_16X16X4_F32.
// One workgroup (256 thr = 8 waves) per (b,h). 12 output tiles (48x64) are
// register-resident: wave w owns tile w, and tile w+8 when w<4.
__global__ void __launch_bounds__(256) kC_attn(
    const float* __restrict__ qg, const float* __restrict__ kg,
    const float* __restrict__ vg, const int* __restrict__ Mtop,
    float* __restrict__ ctx) {
  __shared__ float Qs[UP][D_ + 1];         // selected Q rows   (12.5 KB)
  __shared__ float Ks[CHUNK][D_ + 1];      // staged K chunk    (16.6 KB)
  __shared__ float Vs[CHUNK][D_ + 1];      // staged V chunk    (16.6 KB)
  __shared__ float Sc[UP][CHUNK + 1];      // score/prob chunk  (12.5 KB)
  __shared__ float mrow[UP], srow[UP], crow[UP];
  __shared__ int   qidx[UP];

  int bh   = blockIdx.x;
  int b    = bh / H_, h = bh % H_;
  int tid  = threadIdx.x;
  int lane = tid & 31;
  int wave = tid >> 5;                     // 0..7

  if (tid < UP) {
    qidx[tid] = (tid < NTOP) ? Mtop[bh * 64 + tid] : (L_ + 1); // pad rows never masked
    mrow[tid] = NEG_BIG;
    srow[tid] = 0.f;
  }
  __syncthreads();
  for (int e = tid; e < UP * D_; e += 256) {
    int r = e / D_, d = e % D_;
    float val = 0.f;
    if (r < NTOP) val = qg[(((size_t)b * L_ + qidx[r]) * H_ + h) * D_ + d];
    Qs[r][d] = val;
  }
  __syncthreads();

  // WMMA f32 16x16x4 lane mapping (ISA 7.12.2):
  //  A (16x4):  lane&15 = M row; vgpr j -> K = j + 2*(lane>>4)
  //  B (4x16):  lane&15 = N col; vgpr j -> K = j + 2*(lane>>4)
  //  C/D(16x16):lane&15 = N col; vgpr j -> M = j + 8*(lane>>4)
  const int m16    = lane & 15;
  const int khalf  = (lane >> 4) * 2;
  const int rowoff = (lane >> 4) * 8;

  // register-resident output accumulators
  v8f accA = {};                // tile job = wave        (jobs 0..7)
  v8f accB = {};                // tile job = wave + 8    (jobs 8..11, wave<4)

  auto qk_tile = [&](int job, int key0) {
    int rt = job >> 2, ct = job & 3;
    int key = key0 + ct * 16 + m16;
    v8f acc = {};
#pragma unroll
    for (int dk = 0; dk < D_ / 4; ++dk) {
      int d0 = dk * 4 + khalf;
      v2f a, bf;
      a[0]  = Qs[rt * 16 + m16][d0];
      a[1]  = Qs[rt * 16 + m16][d0 + 1];
      bf[0] = Ks[ct * 16 + m16][d0];
      bf[1] = Ks[ct * 16 + m16][d0 + 1];
      acc = __builtin_amdgcn_wmma_f32_16x16x4_f32(false, a, false, bf,
                                                  (short)0, acc, false, false);
    }
#pragma unroll
    for (int j = 0; j < 8; ++j) {
      int r = rt * 16 + rowoff + j;
      float val = acc[j] * SCALE;
      if (key > qidx[r]) val = -__builtin_inff();   // causal: key_pos > M_top
      Sc[r][ct * 16 + m16] = val;
    }
  };

  auto rescale_tile = [&](v8f& acc, int job) {
    int rt = job >> 2;
#pragma unroll
    for (int j = 0; j < 8; ++j) acc[j] *= crow[rt * 16 + rowoff + j];
  };

  auto pv_tile = [&](v8f& acc, int job) {
    int rt = job >> 2, nt = job & 3;
#pragma unroll
    for (int kq = 0; kq < CHUNK / 4; ++kq) {
      int kk = kq * 4 + khalf;
      v2f a, bf;
      a[0]  = Sc[rt * 16 + m16][kk];
      a[1]  = Sc[rt * 16 + m16][kk + 1];
      bf[0] = Vs[kk][nt * 16 + m16];
      bf[1] = Vs[kk + 1][nt * 16 + m16];
      acc = __builtin_amdgcn_wmma_f32_16x16x4_f32(false, a, false, bf,
                                                  (short)0, acc, false, false);
    }
  };

  for (int ch = 0; ch < NCHUNK; ++ch) {
    const int key0 = ch * CHUNK;

    // prefetch next chunk's K/V rows (global_prefetch_b8)
    if (ch + 1 < NCHUNK && tid < CHUNK) {
      __builtin_prefetch(kg + (((size_t)b * L_ + key0 + CHUNK + tid) * H_ + h) * D_, 0, 0);
      __builtin_prefetch(vg + (((size_t)b * L_ + key0 + CHUNK + tid) * H_ + h) * D_, 0, 0);
    }
    // stage K and V chunk into LDS (coalesced b128 global loads)
    for (int idx = tid; idx < CHUNK * (D_ / 4); idx += 256) {
      int r  = idx >> 4;
      int c4 = (idx & 15) * 4;
      const float* kr = kg + (((size_t)b * L_ + key0 + r) * H_ + h) * D_ + c4;
      const float* vr = vg + (((size_t)b * L_ + key0 + r) * H_ + h) * D_ + c4;
      float4 kv = *(const float4*)kr;
      float4 vv = *(const float4*)vr;
      Ks[r][c4] = kv.x; Ks[r][c4 + 1] = kv.y; Ks[r][c4 + 2] = kv.z; Ks[r][c4 + 3] = kv.w;
      Vs[r][c4] = vv.x; Vs[r][c4 + 1] = vv.y; Vs[r][c4 + 2] = vv.z; Vs[r][c4 + 3] = vv.w;
    }
    __syncthreads();

    // ---- 1) S = scale * Q @ K^T  (12 tile jobs, wave-uniform)
    qk_tile(wave, key0);
    if (wave < 4) qk_tile(wave + 8, key0);
    __syncthreads();

    // ---- 2) online softmax row update (one thread per row)
    if (tid < UP) {
      int r = tid;
      float mc = NEG_BIG;
      for (int c = 0; c < CHUNK; ++c) mc = fmaxf(mc, Sc[r][c]);
      float mold = mrow[r];
      float mn   = fmaxf(mold, mc);
      float corr = __expf(mold - mn);
      float sum  = 0.f;
      for (int c = 0; c < CHUNK; ++c) {
        float e = __expf(Sc[r][c] - mn);
        Sc[r][c] = e;
        sum += e;
      }
      srow[r] = srow[r] * corr + sum;
      mrow[r] = mn;
      crow[r] = corr;
    }
    __syncthreads();

    // ---- 3) rescale register accumulators, then O += P @ V
    rescale_tile(accA, wave);
    if (wave < 4) rescale_tile(accB, wave + 8);
    pv_tile(accA, wave);
    if (wave < 4) pv_tile(accB, wave + 8);
    __syncthreads();   // protect Ks/Vs/Sc before next chunk overwrites
  }

  // ---- finalize: ctx_sel = O / l  (register tiles -> global)
  auto store_tile = [&](v8f& acc, int job) {
    int rt = job >> 2, nt = job & 3;
#pragma unroll
    for (int j = 0; j < 8; ++j) {
      int r = rt * 16 + rowoff + j;
      if (r < NTOP)
        ctx[((size_t)bh * UP + r) * D_ + nt * 16 + m16] = acc[j] / srow[r];
    }
  };
  store_tile(accA, wave);
  if (wave < 4) store_tile(accB, wave + 8);
}

// ---------------------------------------------------------------- cumsum (3 passes)
#define NCH 64
#define CL  64   // NCH*CL == L_

__global__ void kD1_partial(const float* __restrict__ v, float* __restrict__ part) {
  int blk = blockIdx.x;               // bh*NCH + c
  int c  = blk % NCH;
  int bh = blk / NCH;
  int b = bh / H_, h = bh % H_;
  int d = threadIdx.x;                // 0..63 -> coalesced 256B rows
  const float* base = v + (((size_t)b * L_ + c * CL) * H_ + h) * D_ + d;
  float s = 0.f;
  for (int i = 0; i < CL; ++i) s += base[(size_t)i * H_ * D_];
  part[((size_t)bh * NCH + c) * 64 + d] = s;
}

__global__ void kD2_scan(float* __restrict__ part) {
  int t = blockIdx.x * blockDim.x + threadIdx.x;
  if (t >= B_ * H_ * D_) return;
  int bh = t / D_, d = t % D_;
  float run = 0.f;
  for (int c = 0; c < NCH; ++c) {
    size_t idx = ((size_t)bh * NCH + c) * 64 + d;
    float tmp = part[idx];
    part[idx] = run;                  // exclusive prefix of chunk sums
    run += tmp;
  }
}

__global__ void kD3_cumsum(const float* __restrict__ v,
                           const float* __restrict__ part,
                           float* __restrict__ out) {
  int blk = blockIdx.x;
  int c  = blk % NCH;
  int bh = blk / NCH;
  int b = bh / H_, h = bh % H_;
  int d = threadIdx.x;
  float run = part[((size_t)bh * NCH + c) * 64 + d];
  const float* base  = v   + (((size_t)b * L_ + c * CL) * H_ + h) * D_ + d;
  float*       obase = out + (((size_t)b * L_ + c * CL) * H_ + h) * D_ + d;
  for (int i = 0; i < CL; ++i) {
    run += base[(size_t)i * H_ * D_];
    obase[(size_t)i * H_ * D_] = run; // out[b,l,h,d] = inclusive cumsum over l
  }
}

// ---------------------------------------------------------------- scatter
__global__ void kE_scatter(const float* __restrict__ ctx,
                           const int* __restrict__ Mtop,
                           float* __restrict__ out) {
  int bh = blockIdx.x;
  int b = bh / H_, h = bh % H_;
  for (int e = threadIdx.x; e < NTOP * D_; e += 256) {
    int u = e / D_, d = e % D_;
    int qi = Mtop[bh * 64 + u];
    out[(((size_t)b * L_ + qi) * H_ + h) * D_ + d] =
        ctx[((size_t)bh * UP + u) * D_ + d];
  }
}

// ---------------------------------------------------------------- launch
extern "C" void kernel_launch(void* const* d_in, const int* in_sizes, int n_in,
                              void* d_out, int out_size, void* d_ws, size_t ws_size,
                              hipStream_t stream) {
  const float* q    = (const float*)d_in[0];
  const float* k    = (const float*)d_in[1];
  const float* v    = (const float*)d_in[2];
  const int*   samp = (const int*)d_in[4];   // d_in[3] = attn_mask (unused)
  float* out = (float*)d_out;

  char* ws = (char*)d_ws;
  float* Mmet = (float*)ws;                                   // B*H*L        = 65536 f
  int*   Mtop = (int*)(ws + (size_t)65536 * 4);               // B*H*64       =  1024 i
  float* ctx  = (float*)(ws + (size_t)65536 * 4 + 4096);      // B*H*48*64    = 49152 f
  float* part = (float*)(ws + (size_t)65536 * 4 + 4096 + (size_t)49152 * 4); // 65536 f

  kA_metric<<<(B_ * H_ * L_ + 255) / 256, 256, 0, stream>>>(q, k, samp, Mmet);
  kB_topk  <<<B_ * H_, 256, 0, stream>>>(Mmet, Mtop);
  kC_attn  <<<B_ * H_, 256, 0, stream>>>(q, k, v, Mtop, ctx);
  kD1_partial<<<B_ * H_ * NCH, 64, 0, stream>>>(v, part);
  kD2_scan <<<(B_ * H_ * D_ + 255) / 256, 256, 0, stream>>>(part);
  kD3_cumsum<<<B_ * H_ * NCH, 64, 0, stream>>>(v, part, out);
  kE_scatter<<<B_ * H_, 256, 0, stream>>>(ctx, Mtop, out);
}